// GATConvs_34608846471835
// MI455X (gfx1250) — compile-verified
//
#include <hip/hip_runtime.h>
#include <math.h>

// ---------------------------------------------------------------------------
// 2-layer GAT for MI455X (gfx1250, wave32).
// GEMMs use V_WMMA_F32_16X16X4_F32 (full f32 precision; GEMM FLOPs are
// negligible vs the scatter-bound edge phase, and A/B live in the 192MB L2).
// Edge softmax: 3 passes with atomics; per-edge exp cached in workspace.
// ---------------------------------------------------------------------------

typedef float v2f __attribute__((ext_vector_type(2)));
typedef float v8f __attribute__((ext_vector_type(8)));

#define DEV __device__ __forceinline__

DEV float wsum(float v) {
#pragma unroll
  for (int m = 16; m; m >>= 1) v += __shfl_xor(v, m, 32);
  return v;
}

// monotone float <-> uint encoding so atomicMax(uint) == float max
DEV unsigned enc_f(float f) {
  unsigned u = __float_as_uint(f);
  return (u & 0x80000000u) ? ~u : (u | 0x80000000u);
}
DEV float dec_f(unsigned u) {
  return __uint_as_float((u & 0x80000000u) ? (u & 0x7FFFFFFFu) : ~u);
}

// ----------------- x0 = mean over features (F=128) -------------------------
__global__ void k_mean(const float* __restrict__ x, float* __restrict__ out,
                       int NT, int N, int threeN) {
  int wid = (blockIdx.x * blockDim.x + threadIdx.x) >> 5;
  int lane = threadIdx.x & 31;
  if (wid >= NT) return;
  const float* r = x + (size_t)wid * 128;
  float s = r[lane] + r[lane + 32] + r[lane + 64] + r[lane + 96];
  s = wsum(s);
  if (lane == 0) {
    int b = wid / N, n = wid % N;
    out[(size_t)b * threeN + n] = s * (1.0f / 128.0f);
  }
}

// ----------------- GEMM: C[M,256] = A[M,K] @ Bw[K,256] ----------------------
// One wave computes a 16x64 tile (4 x 16x16 accumulators, A-frag reused 4x).
// f32 WMMA fragment layout (ISA 7.12.2): A 16x4 -> 2 VGPRs/lane,
// k_local = vgpr + 2*(lane>=16), M = lane%16.  C/D: M = vgpr + 8*(lane>=16).
__global__ void k_gemm(const float* __restrict__ A, const float* __restrict__ Bw,
                       float* __restrict__ C, int M, int K) {
  int wid = (blockIdx.x * blockDim.x + threadIdx.x) >> 5;
  int lane = threadIdx.x & 31;
  int rowTile = wid >> 2, colGrp = wid & 3;
  int row0 = rowTile << 4, col0 = colGrp << 6;
  if (row0 >= M) return;
  int half = lane >> 4, lm = lane & 15;

  v8f acc0 = {}, acc1 = {}, acc2 = {}, acc3 = {};
  const float* ap = A + (size_t)(row0 + lm) * K + 2 * half;
  const float* bp = Bw + (size_t)(2 * half) * 256 + col0 + lm;

  for (int k0 = 0; k0 < K; k0 += 4) {
    v2f a;
    a.x = ap[k0];
    a.y = ap[k0 + 1];
    const float* b0 = bp + (size_t)k0 * 256;
    v2f b;
    b.x = b0[0];  b.y = b0[256];
    acc0 = __builtin_amdgcn_wmma_f32_16x16x4_f32(false, a, false, b, (short)0, acc0, false, false);
    b.x = b0[16]; b.y = b0[256 + 16];
    acc1 = __builtin_amdgcn_wmma_f32_16x16x4_f32(false, a, false, b, (short)0, acc1, false, false);
    b.x = b0[32]; b.y = b0[256 + 32];
    acc2 = __builtin_amdgcn_wmma_f32_16x16x4_f32(false, a, false, b, (short)0, acc2, false, false);
    b.x = b0[48]; b.y = b0[256 + 48];
    acc3 = __builtin_amdgcn_wmma_f32_16x16x4_f32(false, a, false, b, (short)0, acc3, false, false);
  }

  float* cp = C + (size_t)(row0 + 8 * half) * 256 + col0 + lm;
#pragma unroll
  for (int r2 = 0; r2 < 8; r2++) {
    cp[(size_t)r2 * 256 + 0]  = acc0[r2];
    cp[(size_t)r2 * 256 + 16] = acc1[r2];
    cp[(size_t)r2 * 256 + 32] = acc2[r2];
    cp[(size_t)r2 * 256 + 48] = acc3[r2];
  }
}

// ----------------- per-node attention coefficients --------------------------
// a_src[n,h] = sum_c xp[n,h*64+c]*att_src[h,c]   (same for dst)
__global__ void k_att(const float* __restrict__ xp, const float* __restrict__ att_s,
                      const float* __restrict__ att_d, float* __restrict__ a_s,
                      float* __restrict__ a_d, int NT) {
  int wid = (blockIdx.x * blockDim.x + threadIdx.x) >> 5;
  int lane = threadIdx.x & 31;
  if (wid >= NT) return;
  const float* row = xp + (size_t)wid * 256;
#pragma unroll
  for (int h = 0; h < 4; h++) {
    int c0 = h * 64 + lane;
    float vs = row[c0] * att_s[c0] + row[c0 + 32] * att_s[c0 + 32];
    float vd = row[c0] * att_d[c0] + row[c0 + 32] * att_d[c0 + 32];
    vs = wsum(vs);
    vd = wsum(vd);
    if (lane == 0) {
      a_s[wid * 4 + h] = vs;
      a_d[wid * 4 + h] = vd;
    }
  }
}

// ----------------- init aggregation / softmax state -------------------------
__global__ void k_init(float* __restrict__ agg, float* __restrict__ denom,
                       unsigned* __restrict__ m, int n_agg, int n_small) {
  int i = blockIdx.x * blockDim.x + threadIdx.x;
  if (i < n_agg) agg[i] = 0.0f;
  if (i < n_small) {
    denom[i] = 0.0f;
    m[i] = 0u;  // < enc of any finite float; self-loops ensure every segment hits
  }
}

DEV void edge_sd(int e, int E, const int* __restrict__ src, const int* __restrict__ dst,
                 int& s, int& d) {
  if (e < E) { s = src[e]; d = dst[e]; }
  else       { s = d = e - E; }  // appended self-loops
}

// ----------------- pass 1: segment max of leaky_relu logits -----------------
__global__ void k_edge_max(const int* __restrict__ src, const int* __restrict__ dst,
                           const float* __restrict__ a_s, const float* __restrict__ a_d,
                           unsigned* __restrict__ m, int E, int ET) {
  int idx = blockIdx.x * blockDim.x + threadIdx.x;  // edge*4 + head
  if (idx >= ET * 4) return;
  int e = idx >> 2, h = idx & 3;
  int s, d;
  edge_sd(e, E, src, dst, s, d);
  float l = a_s[s * 4 + h] + a_d[d * 4 + h];
  l = (l > 0.0f) ? l : 0.2f * l;
  atomicMax(&m[d * 4 + h], enc_f(l));
}

// ----------------- pass 2: e = exp(logit - max), denom += e -----------------
__global__ void k_edge_exp(const int* __restrict__ src, const int* __restrict__ dst,
                           const float* __restrict__ a_s, const float* __restrict__ a_d,
                           const unsigned* __restrict__ m, float* __restrict__ ew,
                           float* __restrict__ denom, int E, int ET) {
  int idx = blockIdx.x * blockDim.x + threadIdx.x;
  if (idx >= ET * 4) return;
  int e = idx >> 2, h = idx & 3;
  int s, d;
  edge_sd(e, E, src, dst, s, d);
  float l = a_s[s * 4 + h] + a_d[d * 4 + h];
  l = (l > 0.0f) ? l : 0.2f * l;
  float ev = expf(l - dec_f(m[d * 4 + h]));
  ew[idx] = ev;
  atomicAdd(&denom[d * 4 + h], ev);
}

// ----------------- pass 3: agg[dst] += alpha * xp[src]  (1 wave / edge) -----
__global__ void k_edge_scatter(const int* __restrict__ src, const int* __restrict__ dst,
                               const float* __restrict__ ew, const float* __restrict__ denom,
                               const float* __restrict__ xp, float* __restrict__ agg,
                               int E, int ET) {
  int wid = (blockIdx.x * blockDim.x + threadIdx.x) >> 5;
  int lane = threadIdx.x & 31;
  if (wid >= ET) return;
  int s, d;
  edge_sd(wid, E, src, dst, s, d);
  float al[4];
#pragma unroll
  for (int h = 0; h < 4; h++)
    al[h] = ew[wid * 4 + h] / (denom[d * 4 + h] + 1e-16f);
  const float* xr = xp + (size_t)s * 256;
  float* ar = agg + (size_t)d * 256;
#pragma unroll
  for (int i = 0; i < 8; i++) {
    int ch = lane + i * 32;
    atomicAdd(&ar[ch], xr[ch] * al[ch >> 6]);
  }
}

// ----------------- bias + ELU + pool projection -----------------------------
__global__ void k_epilogue(const float* __restrict__ agg, const float* __restrict__ bias,
                           const float* __restrict__ pw, const float* __restrict__ pb,
                           float* __restrict__ hout, float* __restrict__ out,
                           int NT, int N, int threeN, int sec) {
  int wid = (blockIdx.x * blockDim.x + threadIdx.x) >> 5;
  int lane = threadIdx.x & 31;
  if (wid >= NT) return;
  const float* ar = agg + (size_t)wid * 256;
  float pool = 0.0f;
#pragma unroll
  for (int i = 0; i < 8; i++) {
    int ch = lane + i * 32;
    float v = ar[ch] + bias[ch];
    v = (v > 0.0f) ? v : expm1f(v);  // ELU(alpha=1)
    hout[(size_t)wid * 256 + ch] = v;
    pool += v * pw[ch];
  }
  pool = wsum(pool);
  if (lane == 0) {
    int b = wid / N, n = wid % N;
    out[(size_t)b * threeN + sec * N + n] = pool + pb[0];
  }
}

// ---------------------------------------------------------------------------
extern "C" void kernel_launch(void* const* d_in, const int* in_sizes, int n_in,
                              void* d_out, int out_size, void* d_ws, size_t ws_size,
                              hipStream_t stream) {
  const float* x   = (const float*)d_in[0];
  const int*   ei  = (const int*)d_in[1];
  const float* W1  = (const float*)d_in[2];
  const float* as1 = (const float*)d_in[3];
  const float* ad1 = (const float*)d_in[4];
  const float* b1  = (const float*)d_in[5];
  const float* W2  = (const float*)d_in[6];
  const float* as2 = (const float*)d_in[7];
  const float* ad2 = (const float*)d_in[8];
  const float* b2  = (const float*)d_in[9];
  const float* pw1 = (const float*)d_in[10];
  const float* pb1 = (const float*)d_in[11];
  const float* pw2 = (const float*)d_in[12];
  const float* pb2 = (const float*)d_in[13];
  float* out = (float*)d_out;

  const int NT = 32768, N = 4096, E = 524288;
  const int ET = E + NT;            // + self loops = 557056
  const int threeN = 3 * N;
  const size_t FEAT_BYTES = (size_t)NT * 256 * 4;  // 32 MiB

  char* w = (char*)d_ws;
  float*    xp    = (float*)(w);                    // transformed features
  float*    hb    = (float*)(w + FEAT_BYTES);       // h1 (layer-2 input)
  float*    agg   = (float*)(w + 2 * FEAT_BYTES);   // scatter accumulator
  float*    a_s   = (float*)(w + 3 * FEAT_BYTES);
  float*    a_d   = a_s + (size_t)NT * 4;
  unsigned* m     = (unsigned*)(a_d + (size_t)NT * 4);
  float*    denom = (float*)(m + (size_t)NT * 4);
  float*    ew    = denom + (size_t)NT * 4;         // ET*4 edge exp cache

  const int* srcA = ei;
  const int* dstA = ei + E;

  dim3 blk(256);
  int gWaveNT  = NT / 8;          // 4096 blocks, 1 wave per node
  int gGemm    = (NT / 16) * 4 / 8;   // 1024 blocks
  int gInit    = (NT * 256) / 256;    // 32768 blocks
  int gEdgeEH  = (ET * 4) / 256;      // 8704 blocks (exact)
  int gEdgeW   = ET / 8;              // 69632 blocks (exact)

  // ---- x0 = mean(x, axis=-1) ----
  k_mean<<<gWaveNT, blk, 0, stream>>>(x, out, NT, N, threeN);

  // ================= layer 1 =================
  k_gemm<<<gGemm, blk, 0, stream>>>(x, W1, xp, NT, 128);
  k_att<<<gWaveNT, blk, 0, stream>>>(xp, as1, ad1, a_s, a_d, NT);
  k_init<<<gInit, blk, 0, stream>>>(agg, denom, m, NT * 256, NT * 4);
  k_edge_max<<<gEdgeEH, blk, 0, stream>>>(srcA, dstA, a_s, a_d, m, E, ET);
  k_edge_exp<<<gEdgeEH, blk, 0, stream>>>(srcA, dstA, a_s, a_d, m, ew, denom, E, ET);
  k_edge_scatter<<<gEdgeW, blk, 0, stream>>>(srcA, dstA, ew, denom, xp, agg, E, ET);
  k_epilogue<<<gWaveNT, blk, 0, stream>>>(agg, b1, pw1, pb1, hb, out, NT, N, threeN, 1);

  // ================= layer 2 =================
  k_gemm<<<gGemm, blk, 0, stream>>>(hb, W2, xp, NT, 256);
  k_att<<<gWaveNT, blk, 0, stream>>>(xp, as2, ad2, a_s, a_d, NT);
  k_init<<<gInit, blk, 0, stream>>>(agg, denom, m, NT * 256, NT * 4);
  k_edge_max<<<gEdgeEH, blk, 0, stream>>>(srcA, dstA, a_s, a_d, m, E, ET);
  k_edge_exp<<<gEdgeEH, blk, 0, stream>>>(srcA, dstA, a_s, a_d, m, ew, denom, E, ET);
  k_edge_scatter<<<gEdgeW, blk, 0, stream>>>(srcA, dstA, ew, denom, xp, agg, E, ET);
  // h2 only needed for pooling; dump it over xp (dead after this point)
  k_epilogue<<<gWaveNT, blk, 0, stream>>>(agg, b2, pw2, pb2, xp, out, NT, N, threeN, 2);
}